// LLFullObjectCondensation_66236985638921
// MI455X (gfx1250) — compile-verified
//
#include <hip/hip_runtime.h>
#include <math.h>

typedef __attribute__((ext_vector_type(2))) float v2f;
typedef __attribute__((ext_vector_type(8))) float v8f;

#define K_CL 256
#define EPSF 1e-6f
#define Q_MIN 0.5f

// ---------------- workspace layout (bytes) ----------------
#define OFF_ALPHA   0                       // K_CL * 8 bytes (u64 packed beta|idx)
#define OFF_AX      (2048)                  // K floats
#define OFF_AY      (2048 + 1024)
#define OFF_QA      (2048 + 2048)           // q_alpha * kmf
#define OFF_ANORM   (2048 + 3072)           // |x_alpha|^2
#define OFF_BALPHA  (2048 + 4096)
#define OFF_KMF     (2048 + 5120)
#define OFF_PAYNUM  (2048 + 6144)           // K*4 floats
#define OFF_PAYDEN  (OFF_PAYNUM + 4096)     // K floats
#define OFF_SCAL    (OFF_PAYDEN + 1024)     // [0]=att [1]=rep [2]=noise_num [3]=noise_den
#define WS_BYTES    (OFF_SCAL + 64)

__device__ __forceinline__ float clip_beta(float b) {
    return fminf(fmaxf(b, 1e-4f), 1.f - 1e-4f);
}

// ---------------- zero workspace ----------------
__global__ void k_init(unsigned* ws) {
    int i = blockIdx.x * blockDim.x + threadIdx.x;
    if (i < WS_BYTES / 4) ws[i] = 0u;
}

// ---------------- alpha selection: packed atomicMax per cluster ----------------
__global__ __launch_bounds__(256) void k_alpha(const float* __restrict__ beta,
                                               const int* __restrict__ tidx,
                                               unsigned long long* __restrict__ packed,
                                               int n) {
    int i = blockIdx.x * blockDim.x + threadIdx.x;
    if (i >= n) return;
    int k = tidx[i];
    if (k < 0 || k >= K_CL) return;
    float b = clip_beta(beta[i]);
    unsigned bb = __float_as_uint(b);              // positive floats: bit order == value order
    unsigned long long key = ((unsigned long long)bb << 32) | (unsigned)i;
    atomicMax(&packed[k], key);
}

// ---------------- extract cluster arrays ----------------
__global__ __launch_bounds__(256) void k_extract(const float* __restrict__ beta,
                                                 const float* __restrict__ cc,
                                                 float* __restrict__ ws_f) {
    int k = threadIdx.x;
    unsigned long long p = ((const unsigned long long*)ws_f)[k];   // OFF_ALPHA == 0
    float ax = 0.f, ay = 0.f, qa = 0.f, an = 0.f, ba = 0.f, km = 0.f;
    if (p != 0ull) {
        int idx = (int)(p & 0xffffffffull);
        ax = cc[2 * idx];
        ay = cc[2 * idx + 1];
        float b = clip_beta(beta[idx]);
        float a = atanhf(b);
        qa = a * a + Q_MIN;            // energy_weights == 1
        an = ax * ax + ay * ay;
        ba = b;
        km = 1.f;
    }
    ws_f[OFF_AX / 4 + k] = ax;
    ws_f[OFF_AY / 4 + k] = ay;
    ws_f[OFF_QA / 4 + k] = qa;       // q_alpha * kmf (0 when invalid)
    ws_f[OFF_ANORM / 4 + k] = an;
    ws_f[OFF_BALPHA / 4 + k] = ba;
    ws_f[OFF_KMF / 4 + k] = km;
}

// ---------------- per-point payload + noise ----------------
__global__ __launch_bounds__(256) void k_payload(const float* __restrict__ beta,
                                                 const float* __restrict__ pe,
                                                 const float* __restrict__ ppos,
                                                 const float* __restrict__ pt,
                                                 const float* __restrict__ pid,
                                                 const int* __restrict__ tidx,
                                                 const float* __restrict__ te,
                                                 const float* __restrict__ tpos,
                                                 const float* __restrict__ tt,
                                                 float* __restrict__ ws_f, int n) {
    __shared__ float s_nn, s_nd;
    if (threadIdx.x == 0) { s_nn = 0.f; s_nd = 0.f; }
    __syncthreads();
    int i = blockIdx.x * blockDim.x + threadIdx.x;
    if (i < n) {
        float b = clip_beta(beta[i]);
        int k = tidx[i];
        if (k < 0) {
            atomicAdd(&s_nn, b);        // noise numerator (S_B = 1)
            atomicAdd(&s_nd, 1.f);      // noise denominator
        } else {
            float balpha = ws_f[OFF_BALPHA / 4 + k];
            float kmf    = ws_f[OFF_KMF / 4 + k];
            if (kmf > 0.f && b > 0.1f * balpha) {    // PAYLOAD_REL_THRESHOLD
                // energy (huber)
                float ten = te[i];
                float ste = sqrtf(ten + 0.001f);
                float l   = fabsf(ten - pe[i]) / (ste + 1.0f);   // ENERGY_DEN_OFFSET
                float d   = ste * 2.0f;                           // HUBER_ENERGY_SCALE
                float eloss = (l < d) ? l * l : (d * d + 2.f * d * (l - d));
                // position
                float dx = tpos[2 * i] - ppos[2 * i];
                float dy = tpos[2 * i + 1] - ppos[2 * i + 1];
                float ploss = (dx * dx + dy * dy) * 0.01f;
                // timing
                float dtm = tt[i] * 1e9f - pt[i];
                float tloss = dtm * dtm;
                // classification
                float cs = 0.f;
                #pragma unroll
                for (int c = 0; c < 6; ++c) { float v = pid[6 * i + c]; cs += v * v; }
                float closs = 1e-8f * (cs * (1.f / 6.f));
                float w = b;   // p_w weight
                atomicAdd(&ws_f[OFF_PAYNUM / 4 + 4 * k + 0], w * eloss);
                atomicAdd(&ws_f[OFF_PAYNUM / 4 + 4 * k + 1], w * ploss);
                atomicAdd(&ws_f[OFF_PAYNUM / 4 + 4 * k + 2], w * tloss);
                atomicAdd(&ws_f[OFF_PAYNUM / 4 + 4 * k + 3], w * closs);
                atomicAdd(&ws_f[OFF_PAYDEN / 4 + k], w);
            }
        }
    }
    __syncthreads();
    if (threadIdx.x == 0) {
        if (s_nn != 0.f) atomicAdd(&ws_f[OFF_SCAL / 4 + 2], s_nn);
        if (s_nd != 0.f) atomicAdd(&ws_f[OFF_SCAL / 4 + 3], s_nd);
    }
}

// ---------------- WMMA: attraction + repulsion over N x K pairs ----------------
// One wave handles a 16-point stripe. Clusters are A (M dim), points are B (N dim).
// D[m][n] = x_alpha[m] . x[n]  via V_WMMA_F32_16X16X4_F32 (coords padded 2 -> K=4).
// Branchless inner loop: repulsion over ALL valid clusters, own-cluster term
// corrected once per point after the tile loop (also yields the attraction term).
__global__ __launch_bounds__(256) void k_rep_att(const float* __restrict__ beta,
                                                 const float* __restrict__ cc,
                                                 const int* __restrict__ tidx,
                                                 const float* __restrict__ ws_f,
                                                 float* __restrict__ scal, int n) {
    __shared__ float2 s_xy[K_CL];   // (ax, ay)
    __shared__ float2 s_aq[K_CL];   // (|x_alpha|^2, q_alpha*kmf)
    __shared__ float s_att, s_rep;
    if (threadIdx.x == 0) { s_att = 0.f; s_rep = 0.f; }
    for (int i = threadIdx.x; i < K_CL; i += blockDim.x) {
        s_xy[i] = make_float2(ws_f[OFF_AX / 4 + i], ws_f[OFF_AY / 4 + i]);
        s_aq[i] = make_float2(ws_f[OFF_ANORM / 4 + i], ws_f[OFF_QA / 4 + i]);
    }
    __syncthreads();

    const int wave = threadIdx.x >> 5;
    const int lane = threadIdx.x & 31;
    const int half = lane >> 4;      // 0: carries K=0,1; 1: carries K=2,3 (zero pad)
    const int l16  = lane & 15;      // point column (C/D N index) == cluster row for A

    // per-lane point data (C/D column n == l16 for both lane halves)
    int p = (blockIdx.x * 8 + wave) * 16 + l16;
    bool pv = p < n;
    float px = pv ? cc[2 * p] : 0.f;
    float py = pv ? cc[2 * p + 1] : 0.f;
    float qn = 0.f;
    int   kt = -2;
    if (pv) {
        float b = clip_beta(beta[p]);
        float a = atanhf(b);
        qn = a * a + Q_MIN;
        kt = tidx[p];
    }
    float pn2 = px * px + py * py;
    float hmask = (half == 0) ? 1.f : 0.f;

    // B operand: 4x16, K rows striped across lane halves; K=2,3 are zero pad
    v2f bvec;
    bvec.x = px * hmask;
    bvec.y = py * hmask;

    float att_acc = 0.f, rep_acc = 0.f;

    #pragma unroll
    for (int t = 0; t < K_CL / 16; ++t) {
        // A operand: 16x4, VGPR0 = K0 (lanes0-15)/K2 (lanes16-31), VGPR1 = K1/K3
        float2 axy = s_xy[t * 16 + l16];       // unconditional ds_load_b64
        v2f avec;
        avec.x = axy.x * hmask;
        avec.y = axy.y * hmask;
        v8f c = {};
        c = __builtin_amdgcn_wmma_f32_16x16x4_f32(
                /*neg_a=*/false, avec, /*neg_b=*/false, bvec,
                /*c_mod=*/(short)0, c, /*reuse_a=*/false, /*reuse_b=*/false);
        #pragma unroll
        for (int r = 0; r < 8; ++r) {
            int m = t * 16 + r + 8 * half;        // C/D row: VGPR r -> M=r / M=r+8
            float2 aq = s_aq[m];                  // one ds_load_b64
            float d2 = fmaxf(pn2 + aq.x - 2.f * c[r], 0.f);
            float dist = __builtin_amdgcn_sqrtf(d2 + EPSF);   // raw v_sqrt_f32
            float w = fmaxf(1.f - dist, 0.f);
            rep_acc = fmaf(qn * aq.y, w, rep_acc);
        }
    }

    // Own-cluster correction, once per point (half==0 lanes only):
    // remove its repulsion contribution, add the attraction (Mf) term.
    if (half == 0 && kt >= 0 && kt < K_CL) {
        float2 xy = s_xy[kt];
        float2 aq = s_aq[kt];
        float dx = px - xy.x;
        float dy = py - xy.y;
        float d2 = dx * dx + dy * dy;
        float qq = qn * aq.y;                     // q_alpha*kmf (kmf==1 when kt occupied)
        att_acc = qq * d2;
        rep_acc -= qq * fmaxf(1.f - __builtin_amdgcn_sqrtf(d2 + EPSF), 0.f);
    }

    atomicAdd(&s_att, att_acc);
    atomicAdd(&s_rep, rep_acc);
    __syncthreads();
    if (threadIdx.x == 0) {
        atomicAdd(&scal[0], s_att);
        atomicAdd(&scal[1], s_rep);
    }
}

// ---------------- combine ----------------
__global__ __launch_bounds__(256) void k_final(const float* __restrict__ ws_f,
                                               float* __restrict__ out, int n) {
    __shared__ float s_nobj, s_minb, s_pay;
    int k = threadIdx.x;
    if (k == 0) { s_nobj = 0.f; s_minb = 0.f; s_pay = 0.f; }
    __syncthreads();
    float kmf = ws_f[OFF_KMF / 4 + k];
    float bal = ws_f[OFF_BALPHA / 4 + k];
    float den = ws_f[OFF_PAYDEN / 4 + k] + EPSF;
    float ps = 0.f;
    #pragma unroll
    for (int c = 0; c < 4; ++c) ps += ws_f[OFF_PAYNUM / 4 + 4 * k + c] / den;
    atomicAdd(&s_nobj, kmf);
    atomicAdd(&s_minb, (1.f - bal) * kmf);
    atomicAdd(&s_pay, kmf * ps);
    __syncthreads();
    if (k == 0) {
        const float* sc = ws_f + OFF_SCAL / 4;
        float inv_n = 1.f / (float)n;
        float att   = sc[0] * inv_n;
        float rep   = sc[1] * inv_n;
        float minb  = s_minb / (s_nobj + EPSF);
        float noise = sc[2] / (sc[3] + EPSF);
        float pay   = s_pay / (s_nobj + EPSF);
        out[0] = att + rep + minb + noise + pay;   // all loss weights == 1
    }
}

extern "C" void kernel_launch(void* const* d_in, const int* in_sizes, int n_in,
                              void* d_out, int out_size, void* d_ws, size_t ws_size,
                              hipStream_t stream) {
    const float* pred_beta = (const float*)d_in[0];
    const float* pred_cc   = (const float*)d_in[1];
    const float* pred_e    = (const float*)d_in[2];
    const float* pred_pos  = (const float*)d_in[3];
    const float* pred_t    = (const float*)d_in[4];
    const float* pred_id   = (const float*)d_in[5];
    const int*   t_idx     = (const int*)d_in[6];
    const float* t_e       = (const float*)d_in[7];
    const float* t_pos     = (const float*)d_in[8];
    const float* t_t       = (const float*)d_in[9];
    // d_in[10] (t_pid) and d_in[11] (rowsplits) are unused by the loss

    float* ws_f = (float*)d_ws;
    int n = in_sizes[0];
    int nb = (n + 255) / 256;

    k_init<<<(WS_BYTES / 4 + 255) / 256, 256, 0, stream>>>((unsigned*)d_ws);
    k_alpha<<<nb, 256, 0, stream>>>(pred_beta, t_idx, (unsigned long long*)d_ws, n);
    k_extract<<<1, 256, 0, stream>>>(pred_beta, pred_cc, ws_f);
    k_payload<<<nb, 256, 0, stream>>>(pred_beta, pred_e, pred_pos, pred_t, pred_id,
                                      t_idx, t_e, t_pos, t_t, ws_f, n);
    int nb2 = (n + 127) / 128;    // 8 waves/block * 16 points/wave
    k_rep_att<<<nb2, 256, 0, stream>>>(pred_beta, pred_cc, t_idx, ws_f,
                                       ws_f + OFF_SCAL / 4, n);
    k_final<<<1, 256, 0, stream>>>(ws_f, (float*)d_out, n);
}